// SampleCPPooling_61083024883827
// MI455X (gfx1250) — compile-verified
//
#include <hip/hip_runtime.h>

// ---------------------------------------------------------------------------
// CDNA5 (gfx1250) CP-GraphConv pipeline.
// fp32 WMMA (V_WMMA_F32_16X16X4_F32) for all GEMMs; streaming atomics for
// the segment-mean edge aggregation.
// ---------------------------------------------------------------------------

typedef __attribute__((ext_vector_type(2))) float v2f;
typedef __attribute__((ext_vector_type(8))) float v8f;

#define N0 200000
#define N1 50000
#define N2 10000
#define E1 800000
#define E2 160000
#define IN_F 128
#define HID 128
#define RANK 64
#define OUT_C 64

// ---------------------------------------------------------------------------
// Row-major fp32 GEMM:  C[M,N] = act(A[M,K] @ B[K,N] + bias)
// N, K compile-time so all LDS/global offsets fold into immediates.
// One wave computes a full 16xN strip: A float2 is loaded once per K-step and
// reused by N/16 WMMAs (4-8x less A traffic). B is staged in LDS with K-pairs
// packed adjacent so each lane's B operand is a single ds_load_b64.
// Requires: M % 16 == 0, N % 16 == 0, K % 4 == 0, K*N <= 8192.
// ---------------------------------------------------------------------------
template <int N, int K, int HAS_BIAS, int DO_RELU>
__global__ void __launch_bounds__(256)
wmma_gemm_f32(const float* __restrict__ A, const float* __restrict__ B,
              const float* __restrict__ bias, float* __restrict__ C, int M)
{
    constexpr int NT = N / 16;            // n-tiles per wave strip
    __shared__ float Bs[K * N];           // packed: pair (k, k+1) adjacent

    const int tid = threadIdx.x;
    // Pack B[k][n] -> Bs[((k>>1)*N + n)*2 + (k&1)]
    for (int i = tid; i < K * N; i += 256) {
        const int k = i / N;
        const int n = i - k * N;
        Bs[(((k >> 1) * N) + n) * 2 + (k & 1)] = B[i];
    }
    __syncthreads();

    const int wave   = tid >> 5;
    const int lane   = tid & 31;
    const int mTiles = M >> 4;
    const int m_tile = blockIdx.x * 8 + wave;
    if (m_tile >= mTiles) return;         // wave-uniform exit; EXEC stays full

    const int row16 = lane & 15;          // M-row (A) / N-col (B,C) within tile
    const int hi    = lane >> 4;          // lane half: selects K pair / M half

    const float* Arow = A + (size_t)(m_tile * 16 + row16) * (size_t)K;

    v8f acc[NT];
    #pragma unroll
    for (int nt = 0; nt < NT; ++nt) acc[nt] = (v8f){0.f,0.f,0.f,0.f,0.f,0.f,0.f,0.f};

    #pragma unroll
    for (int k0 = 0; k0 < K; k0 += 4) {
        const int ka = k0 + 2 * hi;       // this lane's first K index (even)
        // A 16x4 layout: lanes 0-15 hold K=k0,k0+1; lanes 16-31 hold K=k0+2,k0+3
        float2 av = *reinterpret_cast<const float2*>(Arow + ka);
        v2f a; a.x = av.x; a.y = av.y;
        #pragma unroll
        for (int nt = 0; nt < NT; ++nt) {
            // B 4x16 layout: VGPR0 -> row ka, VGPR1 -> row ka+1 (packed pair)
            const v2f b = *reinterpret_cast<const v2f*>(
                &Bs[(((ka >> 1) * N) + nt * 16 + row16) * 2]);
            acc[nt] = __builtin_amdgcn_wmma_f32_16x16x4_f32(
                /*neg_a=*/false, a, /*neg_b=*/false, b,
                /*c_mod=*/(short)0, acc[nt], /*reuse_a=*/false, /*reuse_b=*/false);
        }
    }

    // C 16x16 layout: VGPR v -> row (v + 8*hi), col row16
    float* Crow = C + (size_t)(m_tile * 16) * (size_t)N;
    #pragma unroll
    for (int nt = 0; nt < NT; ++nt) {
        const int ncol = nt * 16 + row16;
        const float bc = HAS_BIAS ? bias[ncol] : 0.f;
        #pragma unroll
        for (int v = 0; v < 8; ++v) {
            float val = acc[nt][v] + bc;
            if (DO_RELU) val = fmaxf(val, 0.f);
            Crow[(size_t)(v + 8 * hi) * N + ncol] = val;
        }
    }
}

// ---------------------------------------------------------------------------
// Zero-fill a float range (accumulators must start at 0 every call).
// ---------------------------------------------------------------------------
__global__ void zero_f32(float* __restrict__ p, int n)
{
    int i = blockIdx.x * 256 + threadIdx.x;
    if (i < n) p[i] = 0.f;
}

// ---------------------------------------------------------------------------
// Edge gather + scatter-add: one thread per (edge, channel) of the 64-wide
// messages. Coalesced 256B row gathers; f32 atomics resolve in L2 (agg fits).
// ---------------------------------------------------------------------------
__global__ void edge_scatter(const float* __restrict__ hsrc,
                             const int* __restrict__ src,
                             const int* __restrict__ dst,
                             float* __restrict__ agg,
                             float* __restrict__ cnt, int nE)
{
    long long tid = (long long)blockIdx.x * 256 + threadIdx.x;
    if (tid >= (long long)nE * 64) return;
    const int col = (int)(tid & 63);
    const int e   = (int)(tid >> 6);
    const int s = src[e];
    const int d = dst[e];
    const float v = hsrc[(size_t)s * 64 + col];
    atomicAdd(&agg[(size_t)d * 64 + col], v);
    if (col == 0) atomicAdd(&cnt[d], 1.0f);
}

// ---------------------------------------------------------------------------
// z = (agg / max(cnt,1)) * hdst   (written in place into agg)
// ---------------------------------------------------------------------------
__global__ void cp_combine(float* __restrict__ agg,
                           const float* __restrict__ cnt,
                           const float* __restrict__ hdst, int rows)
{
    int i = blockIdx.x * 256 + threadIdx.x;
    if (i >= rows * 64) return;
    const int r = i >> 6;
    const float c = fmaxf(cnt[r], 1.0f);
    agg[i] = (agg[i] / c) * hdst[i];
}

// ---------------------------------------------------------------------------
// Host-side orchestration (graph-capture safe: only stream launches).
// ---------------------------------------------------------------------------
extern "C" void kernel_launch(void* const* d_in, const int* in_sizes, int n_in,
                              void* d_out, int out_size, void* d_ws, size_t ws_size,
                              hipStream_t stream)
{
    (void)in_sizes; (void)n_in; (void)out_size; (void)ws_size;

    const float* x      = (const float*)d_in[0];
    const float* W_src1 = (const float*)d_in[1];
    const float* W_dst1 = (const float*)d_in[2];
    const float* W_out1 = (const float*)d_in[3];
    const float* b1     = (const float*)d_in[4];
    const float* W_src2 = (const float*)d_in[5];
    const float* W_dst2 = (const float*)d_in[6];
    const float* W_out2 = (const float*)d_in[7];
    const float* b2     = (const float*)d_in[8];
    const int*   src1   = (const int*)d_in[9];
    const int*   dst1   = (const int*)d_in[10];
    const int*   src2   = (const int*)d_in[11];
    const int*   dst2   = (const int*)d_in[12];

    float* ws = (float*)d_ws;
    // workspace layout (floats)
    float* hsrc1 = ws;                         // N0*64  = 12,800,000
    float* agg1  = hsrc1 + (size_t)N0 * 64;    // N1*64  =  3,200,000
    float* cnt1  = agg1  + (size_t)N1 * 64;    // N1     =     50,000
    float* hdst1 = cnt1  + (size_t)N1;         // N1*64  =  3,200,000
    float* h     = hdst1 + (size_t)N1 * 64;    // N1*128 =  6,400,000
    float* hsrc2 = h     + (size_t)N1 * 128;   // N1*64  =  3,200,000
    float* agg2  = hsrc2 + (size_t)N1 * 64;    // N2*64  =    640,000
    float* cnt2  = agg2  + (size_t)N2 * 64;    // N2     =     10,000
    float* hdst2 = cnt2  + (size_t)N2;         // N2*64  =    640,000
    float* out   = (float*)d_out;              // N2*64

    // 0) zero accumulators (agg1+cnt1 contiguous, agg2+cnt2 contiguous)
    {
        int n1 = N1 * 64 + N1;
        int n2 = N2 * 64 + N2;
        zero_f32<<<(n1 + 255) / 256, 256, 0, stream>>>(agg1, n1);
        zero_f32<<<(n2 + 255) / 256, 256, 0, stream>>>(agg2, n2);
    }

    // grid: 8 m-tiles (of 16 rows) per block
    auto blocks_for = [](int M) { return (M / 16 + 7) / 8; };

    // ---- Layer 1 ----
    // hsrc1 = x @ W_src1          (200000 x 64, K=128)
    wmma_gemm_f32<64, 128, 0, 0><<<blocks_for(N0), 256, 0, stream>>>(
        x, W_src1, nullptr, hsrc1, N0);
    // hdst1 = x[:N1] @ W_dst1     (50000 x 64, K=128)
    wmma_gemm_f32<64, 128, 0, 0><<<blocks_for(N1), 256, 0, stream>>>(
        x, W_dst1, nullptr, hdst1, N1);
    // segment-sum over E1 edges
    edge_scatter<<<(int)(((long long)E1 * 64 + 255) / 256), 256, 0, stream>>>(
        hsrc1, src1, dst1, agg1, cnt1, E1);
    // agg1 = (agg1/cnt1) * hdst1
    cp_combine<<<(N1 * 64 + 255) / 256, 256, 0, stream>>>(agg1, cnt1, hdst1, N1);
    // h = relu(agg1 @ W_out1 + b1)  (50000 x 128, K=64)
    wmma_gemm_f32<128, 64, 1, 1><<<blocks_for(N1), 256, 0, stream>>>(
        agg1, W_out1, b1, h, N1);

    // ---- Layer 2 ----
    // hsrc2 = h @ W_src2          (50000 x 64, K=128)
    wmma_gemm_f32<64, 128, 0, 0><<<blocks_for(N1), 256, 0, stream>>>(
        h, W_src2, nullptr, hsrc2, N1);
    // hdst2 = h[:N2] @ W_dst2     (10000 x 64, K=128)
    wmma_gemm_f32<64, 128, 0, 0><<<blocks_for(N2), 256, 0, stream>>>(
        h, W_dst2, nullptr, hdst2, N2);
    // segment-sum over E2 edges
    edge_scatter<<<(int)(((long long)E2 * 64 + 255) / 256), 256, 0, stream>>>(
        hsrc2, src2, dst2, agg2, cnt2, E2);
    // agg2 = (agg2/cnt2) * hdst2
    cp_combine<<<(N2 * 64 + 255) / 256, 256, 0, stream>>>(agg2, cnt2, hdst2, N2);
    // out = agg2 @ W_out2 + b2    (10000 x 64, K=64)
    wmma_gemm_f32<64, 64, 1, 0><<<blocks_for(N2), 256, 0, stream>>>(
        agg2, W_out2, b2, out, N2);
}